// GNN_47115791237147
// MI455X (gfx1250) — compile-verified
//
#include <hip/hip_runtime.h>

typedef __attribute__((ext_vector_type(2))) float v2f;
typedef __attribute__((ext_vector_type(8))) float v8f;

static __device__ __forceinline__ v8f wmma4(v2f a, v2f b, v8f c) {
  // V_WMMA_F32_16X16X4_F32 : D(16x16) = A(16x4) * B(4x16) + C
  return __builtin_amdgcn_wmma_f32_16x16x4_f32(false, a, false, b, (short)0, c,
                                               false, false);
}

// ---------------- degree / symmetric norm ----------------
__global__ void k_deg_init(float* __restrict__ deg, int n) {
  int i = blockIdx.x * blockDim.x + threadIdx.x;
  if (i < n) deg[i] = 1.0f;  // self-loop contribution
}

__global__ void k_deg_edges(const long long* __restrict__ dst, float* deg, int e) {
  int i = blockIdx.x * blockDim.x + threadIdx.x;
  if (i < e) atomicAdd(&deg[(int)dst[i]], 1.0f);
}

__global__ void k_rsqrt_inplace(float* __restrict__ deg, int n) {
  int i = blockIdx.x * blockDim.x + threadIdx.x;
  if (i < n) deg[i] = rsqrtf(deg[i]);  // deg >= 1 always
}

// ---- layer-1 GEMM (fused self-loop init):
//      h1[N,16] = x[N,4] @ W1[4,16];  agg1 = dinv^2 * h1
__global__ void k_gemm1(const float* __restrict__ x, const float* __restrict__ W,
                        const float* __restrict__ dinv, float* __restrict__ h,
                        float* __restrict__ agg, int nTiles) {
  int wave = (int)((blockIdx.x * (unsigned)blockDim.x + threadIdx.x) >> 5);
  if (wave >= nTiles) return;  // wave-uniform: EXEC all-1s at WMMA
  int lane = threadIdx.x & 31;
  int l = lane & 15;           // M (for A) / N column (for B, C, D)
  int half = lane >> 4;        // K-half: lanes 0-15 -> K=0,1 ; 16-31 -> K=2,3
  size_t node = (size_t)wave * 16 + l;

  v2f a;
  { const float* p = x + node * 4 + half * 2; a[0] = p[0]; a[1] = p[1]; }
  v2f b;
  { int r = half * 2; b[0] = W[r * 16 + l]; b[1] = W[(r + 1) * 16 + l]; }
  v8f c = {};
  c = wmma4(a, b, c);

  size_t row = (size_t)wave * 16 + half * 8;
#pragma unroll
  for (int r = 0; r < 8; ++r) {
    float di = dinv[row + r];            // lane-uniform broadcast
    h[(row + r) * 16 + l] = c[r];
    agg[(row + r) * 16 + l] = di * di * c[r];
  }
}

// ---- layer-2 GEMM (fused act-1 on load + self-loop init on store):
//      h1a = relu(agg1 + b1);  h2[N,32] = h1a @ W2[16,32];  agg2 = dinv^2 * h2
__global__ void k_gemm2(const float* __restrict__ aggIn, const float* __restrict__ bIn,
                        const float* __restrict__ W, const float* __restrict__ dinv,
                        float* __restrict__ o, float* __restrict__ aggOut,
                        int nTiles) {
  int wave = (int)((blockIdx.x * (unsigned)blockDim.x + threadIdx.x) >> 5);
  if (wave >= nTiles) return;
  int lane = threadIdx.x & 31;
  int l = lane & 15;
  int half = lane >> 4;
  size_t node = (size_t)wave * 16 + l;

  v8f c0 = {}, c1 = {};
#pragma unroll
  for (int kk = 0; kk < 4; ++kk) {
    int k = kk * 4 + half * 2;
    v2f a;
    {
      const float* p = aggIn + node * 16 + k;
      a[0] = fmaxf(p[0] + bIn[k + 0], 0.0f);  // fused bias + ReLU
      a[1] = fmaxf(p[1] + bIn[k + 1], 0.0f);
    }
    v2f b0, b1;
    b0[0] = W[k * 32 + l];        b0[1] = W[(k + 1) * 32 + l];
    b1[0] = W[k * 32 + 16 + l];   b1[1] = W[(k + 1) * 32 + 16 + l];
    c0 = wmma4(a, b0, c0);
    c1 = wmma4(a, b1, c1);
  }
  size_t row = (size_t)wave * 16 + half * 8;
#pragma unroll
  for (int r = 0; r < 8; ++r) {
    float di = dinv[row + r];
    float w = di * di;
    o[(row + r) * 32 + l]      = c0[r];
    o[(row + r) * 32 + 16 + l] = c1[r];
    aggOut[(row + r) * 32 + l]      = w * c0[r];
    aggOut[(row + r) * 32 + 16 + l] = w * c1[r];
  }
}

// ---- FC (fused act-2 on load):
//      h2a = relu(agg2 + b2);  out[N,4] = h2a @ Wfc[32,4] + bfc
__global__ void k_fc(const float* __restrict__ aggIn, const float* __restrict__ bIn,
                     const float* __restrict__ W, const float* __restrict__ bias,
                     float* __restrict__ out, int nTiles) {
  int wave = (int)((blockIdx.x * (unsigned)blockDim.x + threadIdx.x) >> 5);
  if (wave >= nTiles) return;
  int lane = threadIdx.x & 31;
  int l = lane & 15;
  int half = lane >> 4;
  size_t node = (size_t)wave * 16 + l;

  v8f c = {};
#pragma unroll
  for (int kk = 0; kk < 8; ++kk) {
    int k = kk * 4 + half * 2;
    v2f a;
    {
      const float* p = aggIn + node * 32 + k;
      a[0] = fmaxf(p[0] + bIn[k + 0], 0.0f);  // fused bias + ReLU
      a[1] = fmaxf(p[1] + bIn[k + 1], 0.0f);
    }
    v2f b;  // pad 32x4 weight to 32x16 with zero columns
    b[0] = (l < 4) ? W[k * 4 + l] : 0.0f;
    b[1] = (l < 4) ? W[(k + 1) * 4 + l] : 0.0f;
    c = wmma4(a, b, c);
  }
  if (l < 4) {
    float bb = bias[l];
    size_t row = (size_t)wave * 16 + half * 8;
#pragma unroll
    for (int r = 0; r < 8; ++r) out[(row + r) * 4 + l] = c[r] + bb;
  }
}

// ---------------- edge scatter (dominant cost; L2-resident f32 atomics) ----
template <int F>
__global__ void k_agg_edges(const long long* __restrict__ src,
                            const long long* __restrict__ dst,
                            const float* __restrict__ dinv,
                            const float* __restrict__ h, float* __restrict__ agg,
                            int e) {
  int i = blockIdx.x * blockDim.x + threadIdx.x;
  if (i >= e) return;
  int s = (int)src[i], d = (int)dst[i];
  float w = dinv[s] * dinv[d];
  const float4* hv = (const float4*)(h + (size_t)s * F);
  float* ad = agg + (size_t)d * F;
#pragma unroll
  for (int q = 0; q < F / 4; ++q) {
    float4 v = hv[q];
    atomicAdd(&ad[q * 4 + 0], w * v.x);
    atomicAdd(&ad[q * 4 + 1], w * v.y);
    atomicAdd(&ad[q * 4 + 2], w * v.z);
    atomicAdd(&ad[q * 4 + 3], w * v.w);
  }
}

// ---------------- launch ----------------
extern "C" void kernel_launch(void* const* d_in, const int* in_sizes, int n_in,
                              void* d_out, int out_size, void* d_ws, size_t ws_size,
                              hipStream_t stream) {
  const float* x       = (const float*)d_in[0];
  const long long* ei  = (const long long*)d_in[1];  // int64 edge_index [2,E]
  const float* W1      = (const float*)d_in[2];
  const float* b1      = (const float*)d_in[3];
  const float* W2      = (const float*)d_in[4];
  const float* b2      = (const float*)d_in[5];
  const float* Wfc     = (const float*)d_in[6];
  const float* bfc     = (const float*)d_in[7];
  float* out = (float*)d_out;

  const int N = in_sizes[0] / 4;   // 100000 (divisible by 16)
  const int E = in_sizes[1] / 2;   // 3200000
  const long long* src = ei;       // row 0 = source
  const long long* dst = ei + E;   // row 1 = target

  // workspace (floats): deg/dinv | h1pre | agg1 | h2pre | agg2   (~39 MB)
  float* deg  = (float*)d_ws;
  float* h1   = deg + N;
  float* agg1 = h1 + (size_t)N * 16;
  float* h2   = agg1 + (size_t)N * 16;
  float* agg2 = h2 + (size_t)N * 32;

  const int T = 256;
  auto gr = [](long long n, int t) { return (int)((n + t - 1) / t); };

  // symmetric normalization
  k_deg_init<<<gr(N, T), T, 0, stream>>>(deg, N);
  k_deg_edges<<<gr(E, T), T, 0, stream>>>(dst, deg, E);
  k_rsqrt_inplace<<<gr(N, T), T, 0, stream>>>(deg, N);  // deg -> dinv

  const int nTiles = N / 16;
  const int gemmBlocks = (nTiles + (T / 32) - 1) / (T / 32);

  // layer 1: WMMA GEMM + fused self-loop init, then edge scatter
  k_gemm1<<<gemmBlocks, T, 0, stream>>>(x, W1, deg, h1, agg1, nTiles);
  k_agg_edges<16><<<gr(E, T), T, 0, stream>>>(src, dst, deg, h1, agg1, E);

  // layer 2: fused relu(agg1+b1) -> WMMA GEMM -> fused self-loop init
  k_gemm2<<<gemmBlocks, T, 0, stream>>>(agg1, b1, W2, deg, h2, agg2, nTiles);
  k_agg_edges<32><<<gr(E, T), T, 0, stream>>>(src, dst, deg, h2, agg2, E);

  // FC: fused relu(agg2+b2) -> WMMA GEMM -> bias
  k_fc<<<gemmBlocks, T, 0, stream>>>(agg2, b2, Wfc, bfc, out, nTiles);
}